// ImputationLoss_39075612459451
// MI455X (gfx1250) — compile-verified
//
#include <hip/hip_runtime.h>
#include <hip/hip_bf16.h>
#include <math.h>

// ---------------------------------------------------------------------------
// ImputationLoss for MI455X (gfx1250, wave32).  B=32, L=4096, C=4 -> scalar.
//
// Two launches total:
//   kernel 1 (block-role fused): CE | R2 | U-GEMM (WMMA) | V-pairwise
//   kernel 2: final combine
//
// js[i,j] = 0.5*(U[i,j]+U[j,i]) - V[i,j]
//   U[i,j] = sum_{l,c} (S_il*Y_ilc)*Y_jlc  : 32x32x16384 f32 GEMM, chained
//                                            V_WMMA_F32_16X16X4_F32 (2x2 tiles)
//   V[i,j] = sum_l mask * sum_c m*log(m+e) : rotation schedule d=1..16
//                                            (symmetry halves the log count)
// All transcendentals are raw v_log_f32 / v_exp_f32 (base-2); args are normal
// and positive so no denorm guard is needed; one ln2 scale at the end.
// ---------------------------------------------------------------------------

typedef __attribute__((ext_vector_type(2))) float v2f;
typedef __attribute__((ext_vector_type(8))) float v8f;

#define B_      32
#define L_      4096
#define EPS_    1e-8f
#define JS_EPS_ 1e-7f
#define L2E_    1.4426950408889634f
#define LN2_    0.6931471805599453f

// block-role partition of kernel 1
#define CE_BLOCKS 512
#define R2_BLOCKS 128
#define U_BLOCKS  8
#define V_BLOCKS  64
#define TOTAL_BLOCKS (CE_BLOCKS + R2_BLOCKS + U_BLOCKS + V_BLOCKS)

// workspace layout (floats)
#define WS_CE   0                        // 512
#define WS_R2   512                      // 128
#define WS_V    (512 + 128)              // 64   (log2 units)
#define WS_U    (512 + 128 + 64)         // 8*1024 (log2 units)

static __device__ __forceinline__ float rlog2(float x) { return __builtin_amdgcn_logf(x); }
static __device__ __forceinline__ float rexp2(float x) { return __builtin_amdgcn_exp2f(x); }

// w = exp(-yevo) row-normalized, built into lds[0..1024).  8 threads per row.
static __device__ __forceinline__ void build_w_lds(const float* __restrict__ yevo,
                                                   float* __restrict__ lds, int t) {
    const int i  = t >> 3;
    const int j0 = (t & 7) * 4;
    float e0 = rexp2(-yevo[i * 32 + j0 + 0] * L2E_);
    float e1 = rexp2(-yevo[i * 32 + j0 + 1] * L2E_);
    float e2 = rexp2(-yevo[i * 32 + j0 + 2] * L2E_);
    float e3 = rexp2(-yevo[i * 32 + j0 + 3] * L2E_);
    float s = (e0 + e1) + (e2 + e3);
    s += __shfl_xor(s, 4, 8);
    s += __shfl_xor(s, 2, 8);
    s += __shfl_xor(s, 1, 8);
    float inv = 1.0f / (s + EPS_);
    lds[i * 32 + j0 + 0] = e0 * inv;
    lds[i * 32 + j0 + 1] = e1 * inv;
    lds[i * 32 + j0 + 2] = e2 * inv;
    lds[i * 32 + j0 + 3] = e3 * inv;
    __syncthreads();
}

// ------------------------- fused main kernel -------------------------------
__global__ __launch_bounds__(256) void ilw_main_kernel(const float4* __restrict__ logits,
                                                       const float4* __restrict__ prob,
                                                       const float4* __restrict__ y_true4,
                                                       const float*  __restrict__ y_true,
                                                       const float*  __restrict__ yevo,
                                                       float* __restrict__ ws) {
    __shared__ float smem[8 * 1024];
    const int t    = threadIdx.x;
    const int blk  = blockIdx.x;
    const int lane = t & 31;
    const int warp = t >> 5;

    if (blk < CE_BLOCKS) {
        // ----------------------------- CE ----------------------------------
        const int idx = blk * 256 + t;
        float4 x = logits[idx];
        float4 y = y_true4[idx];
        float mx = fmaxf(fmaxf(x.x, x.y), fmaxf(x.z, x.w));
        float se = rexp2((x.x - mx) * L2E_) + rexp2((x.y - mx) * L2E_) +
                   rexp2((x.z - mx) * L2E_) + rexp2((x.w - mx) * L2E_);  // [1,4]
        float xl = (y.x > 0.5f) ? x.x : ((y.y > 0.5f) ? x.y : ((y.z > 0.5f) ? x.z : x.w));
        float v = -(xl - mx - LN2_ * rlog2(se));

        smem[t] = v;
        __syncthreads();
        #pragma unroll
        for (int off = 128; off > 0; off >>= 1) {
            if (t < off) smem[t] += smem[t + off];
            __syncthreads();
        }
        if (t == 0) ws[WS_CE + blk] = smem[0];

    } else if (blk < CE_BLOCKS + R2_BLOCKS) {
        // ----------------------------- R2 ----------------------------------
        const int idx = (blk - CE_BLOCKS) * 256 + t;   // (g,l)
        const int g = idx >> 12;
        const int l = idx & (L_ - 1);
        float alt = 0.0f;
        float pa[4];
        #pragma unroll
        for (int k = 0; k < 4; ++k) {
            const int e = (g * 4 + k) * L_ + l;
            float4 y = y_true4[e];
            float4 p = prob[e];
            alt += (y.x > 0.5f) ? 0.0f : 1.0f;
            pa[k] = p.y + p.z + p.w;
        }
        float af = alt * 0.25f;
        bool mono = (af == 0.0f) || (af == 1.0f);
        float afe = mono ? 0.5f : af;
        float denom = fmaxf(afe * (1.0f - afe), 0.01f);
        float msq = 0.0f;
        #pragma unroll
        for (int k = 0; k < 4; ++k) { float d = pa[k] - afe; msq = fmaf(d, d, msq); }
        msq *= 0.25f;
        float v = mono ? 0.0f : (msq / denom);

        smem[t] = v;
        __syncthreads();
        #pragma unroll
        for (int off = 128; off > 0; off >>= 1) {
            if (t < off) smem[t] += smem[t + off];
            __syncthreads();
        }
        if (t == 0) ws[WS_R2 + (blk - CE_BLOCKS)] = smem[0];

    } else if (blk < CE_BLOCKS + R2_BLOCKS + U_BLOCKS) {
        // ------------------- U GEMM (32x32x16384, WMMA) ---------------------
        const int ub     = blk - (CE_BLOCKS + R2_BLOCKS);
        const int waveId = ub * 8 + warp;
        const int nW     = U_BLOCKS * 8;            // 64 waves
        const int row0   = lane & 15;
        const int row1   = row0 + 16;
        const int cpair  = (lane < 16) ? 0 : 1;     // K components {0,1} / {2,3}

        v8f a00 = {}, a01 = {}, a10 = {}, a11 = {};
        const float2* y2 = (const float2*)y_true;

        for (int l = waveId; l < L_; l += nW) {     // wave-uniform: EXEC all-1s
            float4 p0 = prob[row0 * L_ + l];
            float4 p1 = prob[row1 * L_ + l];
            float S0 =        p0.x * rlog2(p0.x + JS_EPS_);
            S0 = fmaf(p0.y, rlog2(p0.y + JS_EPS_), S0);
            S0 = fmaf(p0.z, rlog2(p0.z + JS_EPS_), S0);
            S0 = fmaf(p0.w, rlog2(p0.w + JS_EPS_), S0);
            float S1 =        p1.x * rlog2(p1.x + JS_EPS_);
            S1 = fmaf(p1.y, rlog2(p1.y + JS_EPS_), S1);
            S1 = fmaf(p1.z, rlog2(p1.z + JS_EPS_), S1);
            S1 = fmaf(p1.w, rlog2(p1.w + JS_EPS_), S1);

            float2 yb0 = y2[(row0 * L_ + l) * 2 + cpair];
            float2 yb1 = y2[(row1 * L_ + l) * 2 + cpair];
            v2f fB0; fB0.x = yb0.x;      fB0.y = yb0.y;
            v2f fB1; fB1.x = yb1.x;      fB1.y = yb1.y;
            v2f fA0; fA0.x = S0 * yb0.x; fA0.y = S0 * yb0.y;
            v2f fA1; fA1.x = S1 * yb1.x; fA1.y = S1 * yb1.y;

            a00 = __builtin_amdgcn_wmma_f32_16x16x4_f32(false, fA0, false, fB0, (short)0, a00, false, false);
            a01 = __builtin_amdgcn_wmma_f32_16x16x4_f32(false, fA0, false, fB1, (short)0, a01, false, false);
            a10 = __builtin_amdgcn_wmma_f32_16x16x4_f32(false, fA1, false, fB0, (short)0, a10, false, false);
            a11 = __builtin_amdgcn_wmma_f32_16x16x4_f32(false, fA1, false, fB1, (short)0, a11, false, false);
        }

        // scatter D-layout -> (i,j)-linear, block-reduce over 8 waves
        const int jj = lane & 15;
        const int ib = (lane < 16) ? 0 : 8;
        #pragma unroll
        for (int r = 0; r < 8; ++r) {
            smem[warp * 1024 + (ib + r) * 32      + jj     ] = a00[r];
            smem[warp * 1024 + (ib + r) * 32      + jj + 16] = a01[r];
            smem[warp * 1024 + (ib + r + 16) * 32 + jj     ] = a10[r];
            smem[warp * 1024 + (ib + r + 16) * 32 + jj + 16] = a11[r];
        }
        __syncthreads();
        for (int q = t; q < 1024; q += 256) {
            float s = 0.0f;
            #pragma unroll
            for (int wv = 0; wv < 8; ++wv) s += smem[wv * 1024 + q];
            ws[WS_U + ub * 1024 + q] = s;
        }

    } else {
        // ---------------- V pairwise (rotation schedule) --------------------
        const int vb     = blk - (CE_BLOCKS + R2_BLOCKS + U_BLOCKS);
        const int waveId = vb * 8 + warp;
        const int nW     = V_BLOCKS * 8;            // 512 waves -> 8 l each

        build_w_lds(yevo, smem, t);                 // smem[0..1024) = w

        float acc[16];
        #pragma unroll
        for (int d = 0; d < 16; ++d) acc[d] = 0.0f;

        for (int l = waveId; l < L_; l += nW) {
            float4 p = prob[lane * L_ + l];
            float4 y = y_true4[lane * L_ + l];
            float lbl = (y.x > 0.5f) ? 0.0f
                      : ((y.y > 0.5f) ? 1.0f : ((y.z > 0.5f) ? 2.0f : 3.0f));

            #pragma unroll
            for (int d = 1; d <= 16; ++d) {
                const int src = (lane + d) & 31;
                float pix = __shfl(p.x, src, 32);
                float piy = __shfl(p.y, src, 32);
                float piz = __shfl(p.z, src, 32);
                float piw = __shfl(p.w, src, 32);
                float li  = __shfl(lbl, src, 32);
                float m0 = 0.5f * (pix + p.x);
                float m1 = 0.5f * (piy + p.y);
                float m2 = 0.5f * (piz + p.z);
                float m3 = 0.5f * (piw + p.w);
                float tt =        m0 * rlog2(m0 + JS_EPS_);
                tt = fmaf(m1, rlog2(m1 + JS_EPS_), tt);
                tt = fmaf(m2, rlog2(m2 + JS_EPS_), tt);
                tt = fmaf(m3, rlog2(m3 + JS_EPS_), tt);
                float mk = (li == lbl) ? 1.0f : 0.0f;
                acc[d - 1] = fmaf(mk, tt, acc[d - 1]);
            }
        }

        // fold symmetric weights: d in 1..15 -> w_ij + w_ji ; d==16 -> w_ij
        float part = 0.0f;
        #pragma unroll
        for (int d = 1; d <= 16; ++d) {
            const int i = (lane + d) & 31;
            float cw = smem[i * 32 + lane];
            if (d < 16) cw += smem[lane * 32 + i];
            part = fmaf(cw, acc[d - 1], part);
        }
        #pragma unroll
        for (int off = 16; off > 0; off >>= 1) part += __shfl_xor(part, off, 32);

        __syncthreads();                            // done reading w
        if (lane == 0) smem[1024 + warp] = part;
        __syncthreads();
        if (t == 0) {
            float s = 0.0f;
            #pragma unroll
            for (int k = 0; k < 8; ++k) s += smem[1024 + k];
            ws[WS_V + vb] = s;                      // log2 units
        }
    }
}

// ---------------------------- final combine --------------------------------
__global__ __launch_bounds__(256) void ilw_final_kernel(const float* __restrict__ ws,
                                                        const float* __restrict__ yevo,
                                                        float* __restrict__ out) {
    __shared__ float wlds[1024];
    __shared__ float4 red[256];
    const int t = threadIdx.x;

    build_w_lds(yevo, wlds, t);

    const float* cep = ws + WS_CE;
    const float* r2p = ws + WS_R2;
    const float* vp  = ws + WS_V;
    const float* up  = ws + WS_U;

    float sce = 0.0f, sr2 = 0.0f, sv = 0.0f, su = 0.0f;
    for (int k = t; k < CE_BLOCKS; k += 256) sce += cep[k];
    if (t < R2_BLOCKS) sr2 = r2p[t];
    if (t < V_BLOCKS)  sv  = vp[t];
    #pragma unroll
    for (int c = 0; c < 4; ++c) {                 // Uterm over 1024 (i,j)
        const int q = t * 4 + c;
        const int i = q >> 5, j = q & 31;
        if (i != j) {
            float uq = 0.0f;
            #pragma unroll
            for (int b = 0; b < U_BLOCKS; ++b) uq += up[b * 1024 + q];
            su = fmaf(0.5f * (wlds[i * 32 + j] + wlds[j * 32 + i]), uq, su);
        }
    }

    red[t] = make_float4(sce, sr2, sv, su);
    __syncthreads();
    #pragma unroll
    for (int off = 128; off > 0; off >>= 1) {
        if (t < off) {
            float4 a = red[t], b = red[t + off];
            red[t] = make_float4(a.x + b.x, a.y + b.y, a.z + b.z, a.w + b.w);
        }
        __syncthreads();
    }
    if (t == 0) {
        float4 s = red[0];
        float ce      = s.x;
        float r2_loss = -s.y * 0.125f;            // -sum(r2)*GROUP/B
        float evo     = LN2_ * (s.w - s.z);       // ln2*(Uterm2 - V2)
        evo = isfinite(evo) ? evo : 0.0f;
        out[0] = ce + r2_loss + evo;
    }
}

// ---------------------------------------------------------------------------
extern "C" void kernel_launch(void* const* d_in, const int* in_sizes, int n_in,
                              void* d_out, int out_size, void* d_ws, size_t ws_size,
                              hipStream_t stream) {
    const float* logits = (const float*)d_in[0];
    const float* prob   = (const float*)d_in[1];
    const float* y_true = (const float*)d_in[2];
    const float* y_evo  = (const float*)d_in[3];
    float* out = (float*)d_out;
    float* ws  = (float*)d_ws;

    ilw_main_kernel<<<TOTAL_BLOCKS, 256, 0, stream>>>(
        (const float4*)logits, (const float4*)prob, (const float4*)y_true,
        y_true, y_evo, ws);
    ilw_final_kernel<<<1, 256, 0, stream>>>(ws, y_evo, out);
}